// GNNLayer_3023656977047
// MI455X (gfx1250) — compile-verified
//
#include <hip/hip_runtime.h>
#include <hip/hip_bf16.h>

#define IN_DIM  128
#define OUT_DIM 128
#define LDSK    132   // padded K-stride for W^T in LDS -> conflict-free ds_load_b64 per half-wave

typedef __attribute__((ext_vector_type(2))) float v2f;
typedef __attribute__((ext_vector_type(4))) float v4f;
typedef __attribute__((ext_vector_type(8))) float v8f;

__global__ __launch_bounds__(256) void zero_ws_kernel(float* __restrict__ p, int n) {
    int i = blockIdx.x * blockDim.x + threadIdx.x;
    if (i < n) p[i] = 0.0f;
}

__global__ __launch_bounds__(256) void degree_kernel(const int* __restrict__ src,
                                                     const int* __restrict__ dst,
                                                     float* __restrict__ deg_out,
                                                     float* __restrict__ deg_in,
                                                     int E) {
    int e = blockIdx.x * blockDim.x + threadIdx.x;
    if (e < E) {
        __hip_atomic_fetch_add(&deg_out[src[e]], 1.0f, __ATOMIC_RELAXED, __HIP_MEMORY_SCOPE_AGENT);
        __hip_atomic_fetch_add(&deg_in[dst[e]],  1.0f, __ATOMIC_RELAXED, __HIP_MEMORY_SCOPE_AGENT);
    }
}

__global__ __launch_bounds__(256) void rinv_kernel(float* __restrict__ deg, int n) {
    int i = blockIdx.x * blockDim.x + threadIdx.x;
    if (i < n) deg[i] = rsqrtf(fmaxf(deg[i], 1.0f));
}

// One wave32 per edge: lanes cover 128 features with float4 (512B coalesced read per edge),
// scaled by rsqrt(out_deg[src]) and atomically accumulated into agg[dst] (L2-resident).
__global__ __launch_bounds__(256) void scatter_kernel(const float* __restrict__ feat,
                                                      const int* __restrict__ src,
                                                      const int* __restrict__ dst,
                                                      const float* __restrict__ rinv_out,
                                                      float* __restrict__ agg,
                                                      int E) {
    int wave = (blockIdx.x * blockDim.x + threadIdx.x) >> 5;
    int lane = threadIdx.x & 31;
    if (wave >= E) return;
    int s = src[wave];
    int d = dst[wave];
    float sc = rinv_out[s];
    v4f v = *(const v4f*)(feat + (size_t)s * IN_DIM + lane * 4);
    float* o = agg + (size_t)d * IN_DIM + lane * 4;
    __hip_atomic_fetch_add(o + 0, v.x * sc, __ATOMIC_RELAXED, __HIP_MEMORY_SCOPE_AGENT);
    __hip_atomic_fetch_add(o + 1, v.y * sc, __ATOMIC_RELAXED, __HIP_MEMORY_SCOPE_AGENT);
    __hip_atomic_fetch_add(o + 2, v.z * sc, __ATOMIC_RELAXED, __HIP_MEMORY_SCOPE_AGENT);
    __hip_atomic_fetch_add(o + 3, v.w * sc, __ATOMIC_RELAXED, __HIP_MEMORY_SCOPE_AGENT);
}

// out = relu( (agg * rsqrt(in_deg)[row]) @ W + b ), via V_WMMA_F32_16X16X4_F32.
// Block = 8 waves; each wave produces one 16-row x 128-col output strip.
__global__ __launch_bounds__(256) void gemm_wmma_kernel(const float* __restrict__ agg,
                                                        const float* __restrict__ W,
                                                        const float* __restrict__ bias,
                                                        const float* __restrict__ rinv_in,
                                                        float* __restrict__ out,
                                                        int nrows, int mtiles) {
    __shared__ float ldsWT[OUT_DIM * LDSK]; // ldsWT[n*LDSK + k] = W[k][n]

    int tid = threadIdx.x;
    // Cooperative transpose-load of W (128x128 f32) into LDS.
    for (int idx = tid; idx < IN_DIM * OUT_DIM; idx += 256) {
        int k = idx >> 7;
        int n = idx & 127;
        ldsWT[n * LDSK + k] = W[idx];
    }
    __syncthreads();

    int wave  = tid >> 5;
    int mtile = blockIdx.x * 8 + wave;
    if (mtile >= mtiles) return;            // whole-wave exit: EXEC stays all-1s for WMMA

    int lane  = tid & 31;
    int nlane = lane & 15;                  // N within 16-wide tile / M row this lane supplies
    int koff  = (lane >> 4) << 1;           // lanes 0-15: K=k,k+1 ; lanes 16-31: K=k+2,k+3

    int row  = mtile * 16 + nlane;          // A-matrix row this lane supplies
    int rowc = row < nrows ? row : (nrows - 1);
    float scale = rinv_in[rowc];
    const float* arow = agg + (size_t)rowc * IN_DIM;

    float bn[8];
#pragma unroll
    for (int nt = 0; nt < 8; ++nt) bn[nt] = bias[nt * 16 + nlane];

    v8f acc[8] = {};

#pragma unroll 2
    for (int kk = 0; kk < IN_DIM; kk += 4) {
        v2f a = *(const v2f*)(arow + kk + koff);
        a.x *= scale;
        a.y *= scale;
#pragma unroll
        for (int nt = 0; nt < 8; ++nt) {
            v2f bf = *(const v2f*)(&ldsWT[(nt * 16 + nlane) * LDSK + kk + koff]);
            acc[nt] = __builtin_amdgcn_wmma_f32_16x16x4_f32(
                false, a, false, bf, (short)0, acc[nt], false, false);
        }
    }

    // D layout: VGPR i -> M = i (lanes 0-15) / i+8 (lanes 16-31); N = nt*16 + nlane.
    int mo = (lane >> 4) << 3;
    int rbase = mtile * 16 + mo;
    if (mtile * 16 + 16 <= nrows) {
        // Full tile (always the case when nrows % 16 == 0): unconditional stores off one
        // base address; all offsets are compile-time immediates within the 24-bit IOFFSET.
        float* obase = out + (size_t)rbase * OUT_DIM + nlane;
#pragma unroll
        for (int nt = 0; nt < 8; ++nt) {
#pragma unroll
            for (int i = 0; i < 8; ++i) {
                float v = acc[nt][i] + bn[nt];
                obase[i * OUT_DIM + nt * 16] = fmaxf(v, 0.0f);
            }
        }
    } else {
#pragma unroll
        for (int nt = 0; nt < 8; ++nt) {
#pragma unroll
            for (int i = 0; i < 8; ++i) {
                int r = rbase + i;
                if (r < nrows) {
                    float v = acc[nt][i] + bn[nt];
                    out[(size_t)r * OUT_DIM + nt * 16 + nlane] = fmaxf(v, 0.0f);
                }
            }
        }
    }
}

extern "C" void kernel_launch(void* const* d_in, const int* in_sizes, int n_in,
                              void* d_out, int out_size, void* d_ws, size_t ws_size,
                              hipStream_t stream) {
    const float* feature = (const float*)d_in[0];
    const float* W       = (const float*)d_in[1];
    const float* b       = (const float*)d_in[2];
    const int*   src     = (const int*)d_in[3];
    const int*   dst     = (const int*)d_in[4];
    float* out = (float*)d_out;

    int N = in_sizes[0] / IN_DIM;   // 50000
    int E = in_sizes[3];            // 600000

    // Workspace layout: agg [N*128] | deg_out [N] | deg_in [N]  (~26 MB)
    float* agg     = (float*)d_ws;
    float* deg_out = agg + (size_t)N * IN_DIM;
    float* deg_in  = deg_out + N;

    int zn = N * IN_DIM + 2 * N;
    zero_ws_kernel<<<(zn + 255) / 256, 256, 0, stream>>>(agg, zn);

    degree_kernel<<<(E + 255) / 256, 256, 0, stream>>>(src, dst, deg_out, deg_in, E);

    // rsqrt(clamp(deg,1)) in place for both arrays (contiguous 2N floats)
    rinv_kernel<<<(2 * N + 255) / 256, 256, 0, stream>>>(deg_out, 2 * N);

    // one wave per edge -> 8 edges per 256-thread block
    scatter_kernel<<<(E + 7) / 8, 256, 0, stream>>>(feature, src, dst, deg_out, agg, E);

    int mtiles = (N + 15) / 16;     // 3125
    gemm_wmma_kernel<<<(mtiles + 7) / 8, 256, 0, stream>>>(agg, W, b, deg_in, out, N, mtiles);
}